// GNNLSTMHybrid_2070174236908
// MI455X (gfx1250) — compile-verified
//
#include <hip/hip_runtime.h>
#include <hip/hip_bf16.h>

typedef float v2f __attribute__((ext_vector_type(2)));
typedef float v8f __attribute__((ext_vector_type(8)));

#define D_IN  128
#define D_HID 256
#define D_OUT 3

// ---------------------------------------------------------------------------
// Kernel 0: zero the scatter accumulators (summed[N*128] ++ counts[N]).
// ---------------------------------------------------------------------------
__global__ void gnn_init_kernel(float* __restrict__ ws, long long n) {
    long long i = (long long)blockIdx.x * blockDim.x + threadIdx.x;
    long long stride = (long long)gridDim.x * blockDim.x;
    for (; i < n; i += stride) ws[i] = 0.0f;
}

// ---------------------------------------------------------------------------
// Kernel 1: fold the two linear layers (no nonlinearity between them):
//   Wc[n][k] = sum_h W2[n][h] * W1[h][k]      (3 x 128)
//   bc[n]    = sum_h W2[n][h] * b1[h] + b2[n] (3)
// Then pre-pack Wc^T into the V_WMMA_F32_16X16X4_F32 B-fragment layout:
//   B is 4x16; VGPR j holds rows {4s+j, 4s+j+2} in lane halves:
//     packedB[s*64 + j*32 + lane] = B[4s + j + (lane>=16 ? 2 : 0)][lane&15]
//   with B[k][n] = Wc[n][k], zero for n >= 3 (N padded to 16).
// ---------------------------------------------------------------------------
__global__ void gnn_fusew_kernel(const float* __restrict__ W1,
                                 const float* __restrict__ b1,
                                 const float* __restrict__ W2,
                                 const float* __restrict__ b2,
                                 float* __restrict__ packedB,
                                 float* __restrict__ bc) {
    __shared__ float Wc[D_OUT][D_IN];
    int tid = threadIdx.x;
    if (tid < D_OUT * D_IN) {
        int n = tid / D_IN, k = tid % D_IN;
        float acc = 0.0f;
        for (int h = 0; h < D_HID; ++h)
            acc += W2[n * D_HID + h] * W1[h * D_IN + k];
        Wc[n][k] = acc;
    }
    if (tid < 4) {
        float acc = 0.0f;
        if (tid < D_OUT) {
            acc = b2[tid];
            for (int h = 0; h < D_HID; ++h)
                acc += W2[tid * D_HID + h] * b1[h];
        }
        bc[tid] = acc;
    }
    __syncthreads();
    for (int idx = tid; idx < 2048; idx += blockDim.x) {
        int s    = idx >> 6;
        int j    = (idx >> 5) & 1;
        int lane = idx & 31;
        int n    = lane & 15;
        int k    = 4 * s + j + ((lane >= 16) ? 2 : 0);
        packedB[idx] = (n < D_OUT) ? Wc[n][k] : 0.0f;
    }
}

// ---------------------------------------------------------------------------
// Kernel 2: scatter-sum. One wave32 per edge: each lane moves one float4 of
// the 128-float source row (512B coalesced), 4 fp32 atomics into summed[dst].
// Working set (25.6MB src + 25.6MB dst) lives in the 192MB L2 -> atomics and
// gathers resolve at L2, no HBM round trips.
// ---------------------------------------------------------------------------
__global__ void gnn_scatter_kernel(const float* __restrict__ x,
                                   const int* __restrict__ edges,
                                   float* __restrict__ summed,
                                   float* __restrict__ counts,
                                   int E) {
    int wave = (int)((blockIdx.x * blockDim.x + threadIdx.x) >> 5);
    int lane = threadIdx.x & 31;
    if (wave >= E) return;
    int src = edges[wave];        // edge_index[0][e]
    int dst = edges[E + wave];    // edge_index[1][e]
    const float4* xr = (const float4*)(x + (long long)src * D_IN);
    float4 v = xr[lane];
    float* srow = summed + (long long)dst * D_IN + lane * 4;
    atomicAdd(srow + 0, v.x);
    atomicAdd(srow + 1, v.y);
    atomicAdd(srow + 2, v.z);
    atomicAdd(srow + 3, v.w);
    if (lane == 0) atomicAdd(counts + dst, 1.0f);
}

// ---------------------------------------------------------------------------
// Kernel 3: fused mean + folded MLP via fp32 WMMA.
// One wave per 16-node tile. A-fragment of V_WMMA_F32_16X16X4_F32:
//   lane L holds row M = L&15; VGPR0 = K (=4s or 4s+2 per lane half),
//   VGPR1 = K+1. Loaded as an 8-byte float2 and scaled by 1/max(count,1).
// 32 chained WMMAs accumulate the 16x16 (N padded) output in 8 VGPRs.
// C/D layout: VGPR j, lanes 0-15 -> M=j, lanes 16-31 -> M=j+8, N=lane&15.
// ---------------------------------------------------------------------------
__global__ void gnn_mlp_kernel(const float* __restrict__ summed,
                               const float* __restrict__ counts,
                               const float* __restrict__ packedB,
                               const float* __restrict__ bc,
                               float* __restrict__ out,
                               int N) {
    int wave = (int)((blockIdx.x * blockDim.x + threadIdx.x) >> 5);
    int lane = threadIdx.x & 31;
    int nTiles = N >> 4;
    if (wave >= nTiles) return;          // wave-uniform: EXEC stays all-1s

    int m0   = wave << 4;
    int mrow = m0 + (lane & 15);
    float inv = 1.0f / fmaxf(counts[mrow], 1.0f);
    int koff = (lane >= 16) ? 2 : 0;
    const float* arow = summed + (long long)mrow * D_IN + koff;

    v8f c = {};
    #pragma unroll 4
    for (int s = 0; s < 32; ++s) {
        float2 a2 = *(const float2*)(arow + 4 * s);   // K = 4s+koff, 4s+koff+1
        v2f a; a[0] = a2.x * inv; a[1] = a2.y * inv;
        v2f b; b[0] = packedB[s * 64 + lane];
               b[1] = packedB[s * 64 + 32 + lane];
        c = __builtin_amdgcn_wmma_f32_16x16x4_f32(
                /*neg_a=*/false, a, /*neg_b=*/false, b,
                /*c_mod=*/(short)0, c, /*reuse_a=*/false, /*reuse_b=*/false);
    }

    int n = lane & 15;
    if (n < D_OUT) {                      // only 3 live output columns
        float bcn = bc[n];
        int rbase = m0 + ((lane >= 16) ? 8 : 0);
        #pragma unroll
        for (int j = 0; j < 8; ++j)
            out[(long long)(rbase + j) * D_OUT + n] = c[j] + bcn;
    }
}

// ---------------------------------------------------------------------------
// Host-side launch. ws layout (floats):
//   [0, N*128)            summed accumulator
//   [N*128, N*129)        counts
//   [N*129, N*129+2048)   packed B fragments (32 K-steps x 2 VGPRs x 32 lanes)
//   [+2048, +2052)        folded bias bc
// ---------------------------------------------------------------------------
extern "C" void kernel_launch(void* const* d_in, const int* in_sizes, int n_in,
                              void* d_out, int out_size, void* d_ws, size_t ws_size,
                              hipStream_t stream) {
    const float* x     = (const float*)d_in[0];
    const int*   edges = (const int*)  d_in[1];   // [2, E] int32 (JAX x64 off)
    const float* W1    = (const float*)d_in[2];   // [256,128]
    const float* b1    = (const float*)d_in[3];   // [256]
    const float* W2    = (const float*)d_in[4];   // [3,256]
    const float* b2    = (const float*)d_in[5];   // [3]
    float*       out   = (float*)d_out;           // [N,3]

    int N = in_sizes[0] / D_IN;   // 50000
    int E = in_sizes[1] / 2;      // 800000

    float* ws = (float*)d_ws;
    long long offSum = 0;
    long long offCnt = (long long)N * D_IN;
    long long offB   = offCnt + N;
    long long offBc  = offB + 2048;

    // 1) zero accumulators (summed + counts)
    gnn_init_kernel<<<4096, 256, 0, stream>>>(ws + offSum, offB);

    // 2) fold W2@W1 / W2@b1+b2 and pack B fragments (independent of step 1)
    gnn_fusew_kernel<<<1, 384, 0, stream>>>(W1, b1, W2, b2, ws + offB, ws + offBc);

    // 3) edge scatter: one wave per edge -> 8 edges per 256-thread block
    int blocksScatter = (E + 7) / 8;
    gnn_scatter_kernel<<<blocksScatter, 256, 0, stream>>>(
        x, edges, ws + offSum, ws + offCnt, E);

    // 4) mean + folded MLP via fp32 WMMA: one wave per 16-node tile
    int nTiles = N / 16;                      // 3125
    int blocksMlp = (nTiles + 7) / 8;         // 8 waves per block
    gnn_mlp_kernel<<<blocksMlp, 256, 0, stream>>>(
        ws + offSum, ws + offCnt, ws + offB, ws + offBc, out, N);
}